// DeepFM_90958817394882
// MI455X (gfx1250) — compile-verified
//
#include <hip/hip_runtime.h>

typedef __attribute__((ext_vector_type(2))) float v2f;
typedef __attribute__((ext_vector_type(8))) float v8f;

#define Bn   65536
#define Fn   39
#define Vn   100000
#define En   16
#define K1   624      // F*E
#define N1   512
#define N2   256
#define EPS  1e-5f
#define MT   128              // M rows per workgroup
#define NMT  (Bn / MT)        // 512 workgroups
#define SBT  18               // transposed Bs stride [n][k]: 16 k + 2 pad

__device__ __forceinline__ v8f wmma4(v2f a, v2f b, v8f c) {
  // V_WMMA_F32_16X16X4_F32 : D = A(16x4) * B(4x16) + C(16x16)
  return __builtin_amdgcn_wmma_f32_16x16x4_f32(
      false, a, false, b, (short)0, c, false, false);
}

// ---------------------------------------------------------------------------
// Kernel 1: fused gather + FM terms + H1 = fm2 @ W1 + b1 + BN1 stat partials.
// 1024 thr = 32 waves; tile 128(M) x 512(fullN); K stepped 16 (one feature f).
// Deep features never touch HBM: gathered straight into LDS A-tiles.
// ---------------------------------------------------------------------------
__global__ __launch_bounds__(1024) void gemm1_fused_kernel(
    const int* __restrict__ Xi, const float* __restrict__ Xv,
    const float* __restrict__ emb1, const float* __restrict__ emb2,
    const float* __restrict__ W1, const float* __restrict__ b1,
    float* __restrict__ H1, float* __restrict__ fs,
    float* __restrict__ cs, float* __restrict__ css)
{
  __shared__ __align__(16) char smem[(MT * 18 + N1 * SBT) * 4]; // 46 KB
  float* As = (float*)smem;                       // [128*18] row-major [m][k]
  float* Bs = (float*)(smem + MT * 18 * 4);       // [512*18] transposed [n][k]
  float* cP  = (float*)smem;                      // alias: [8*512] (post-loop)
  float* cPS = (float*)(smem + 8 * N1 * 4);       // alias: [8*512]

  const int tid  = threadIdx.x;
  const int wave = tid >> 5, lane = tid & 31;
  const int l16  = lane & 15, half = lane >> 4;
  const int waveM = wave & 7;        // M slot: rows waveM*16..+15
  const int waveN = wave >> 3;       // N slot: cols waveN*128..+127 (8 tiles)
  const int mBase = blockIdx.x * MT;

  v8f acc[8] = {};

  // staging roles
  const bool stA = (tid < 512);
  const int sr = tid >> 2;           // A-stage row 0..127
  const int sq = tid & 3;            // A-stage e-quad 0..3
  const int bk  = tid >> 6;          // B-stage k row 0..15
  const int bnq = tid & 63;          // B-stage n-quad

  float s4[4]  = {0.f, 0.f, 0.f, 0.f};   // per-(row,e) FM sums
  float ss4[4] = {0.f, 0.f, 0.f, 0.f};
  float firstacc = 0.f;

  #pragma unroll 1
  for (int f = 0; f < Fn; ++f) {
    __syncthreads();
    if (stA) {
      int   id = Xi[(size_t)(mBase + sr) * Fn + f];
      float xv = Xv[(size_t)(mBase + sr) * Fn + f];
      float4 v = *(const float4*)(emb2 + ((size_t)f * Vn + id) * En + sq * 4);
      float e0 = v.x * xv, e1 = v.y * xv, e2 = v.z * xv, e3 = v.w * xv;
      float* dst = &As[sr * 18 + sq * 4];
      dst[0] = e0; dst[1] = e1; dst[2] = e2; dst[3] = e3;
      s4[0] += e0; ss4[0] += e0 * e0;
      s4[1] += e1; ss4[1] += e1 * e1;
      s4[2] += e2; ss4[2] += e2 * e2;
      s4[3] += e3; ss4[3] += e3 * e3;
      if (sq == 0) firstacc += emb1[(size_t)f * Vn + id] * xv;
    }
    {
      const float* wrow = W1 + (size_t)(f * 16 + bk) * N1;
      float4 w0 = *(const float4*)(wrow + bnq * 4);
      float4 w1 = *(const float4*)(wrow + 256 + bnq * 4);
      const int n0 = bnq * 4;
      Bs[(n0 + 0) * SBT + bk] = w0.x;
      Bs[(n0 + 1) * SBT + bk] = w0.y;
      Bs[(n0 + 2) * SBT + bk] = w0.z;
      Bs[(n0 + 3) * SBT + bk] = w0.w;
      Bs[(n0 + 256) * SBT + bk] = w1.x;
      Bs[(n0 + 257) * SBT + bk] = w1.y;
      Bs[(n0 + 258) * SBT + bk] = w1.z;
      Bs[(n0 + 259) * SBT + bk] = w1.w;
    }
    __syncthreads();
    const int mL = waveM * 16 + l16;
    #pragma unroll
    for (int kk = 0; kk < 4; ++kk) {
      const int kr = kk * 4 + half * 2;
      v2f a = *(const v2f*)(&As[mL * 18 + kr]);
      #pragma unroll
      for (int j = 0; j < 8; ++j) {
        const int col = waveN * 128 + j * 16 + l16;
        v2f b = *(const v2f*)(&Bs[col * SBT + kr]);
        acc[j] = wmma4(a, b, acc[j]);
      }
    }
  }
  __syncthreads();   // K loop done; As/Bs dead, cP/cPS alias them

  // FM first + second order per row (reduce the 4 e-quad lanes)
  if (stA) {
    float p = firstacc;
    #pragma unroll
    for (int i = 0; i < 4; ++i) p += 0.5f * (s4[i] * s4[i] - ss4[i]);
    p += __shfl_xor(p, 1, 32);
    p += __shfl_xor(p, 2, 32);
    if (sq == 0) fs[mBase + sr] = p;
  }

  // H1 store + per-column sum/sumsq partials
  const int rowG = mBase + waveM * 16 + half * 8;
  #pragma unroll
  for (int j = 0; j < 8; ++j) {
    const int col = waveN * 128 + j * 16 + l16;
    float bv = b1[col];
    float s = 0.f, ss = 0.f;
    #pragma unroll
    for (int v = 0; v < 8; ++v) {
      float h = acc[j][v] + bv;
      H1[(size_t)(rowG + v) * N1 + col] = h;
      s += h; ss += h * h;
    }
    s  += __shfl_xor(s, 16, 32);
    ss += __shfl_xor(ss, 16, 32);
    if (half == 0) { cP[waveM * N1 + col] = s; cPS[waveM * N1 + col] = ss; }
  }
  __syncthreads();
  if (tid < N1) {
    float s = 0.f, ss = 0.f;
    #pragma unroll
    for (int w = 0; w < 8; ++w) { s += cP[w * N1 + tid]; ss += cPS[w * N1 + tid]; }
    cs [(size_t)blockIdx.x * N1 + tid] = s;
    css[(size_t)blockIdx.x * N1 + tid] = ss;
  }
}

// ---------------------------------------------------------------------------
// Kernel 2: reduce BN1 partials -> a1 = g1*rsqrt(var+eps), c1 = be1 - mu*a1
// ---------------------------------------------------------------------------
__global__ __launch_bounds__(128) void fin1_kernel(
    const float* __restrict__ cs, const float* __restrict__ css,
    const float* __restrict__ g1, const float* __restrict__ be1,
    float* __restrict__ a1, float* __restrict__ c1)
{
  int k = blockIdx.x * 128 + threadIdx.x;
  float s = 0.f, ss = 0.f;
  for (int mt = 0; mt < NMT; ++mt) { s += cs[(size_t)mt * N1 + k]; ss += css[(size_t)mt * N1 + k]; }
  float mu  = s  * (1.0f / Bn);
  float var = ss * (1.0f / Bn) - mu * mu;
  float a = g1[k] / sqrtf(var + EPS);
  a1[k] = a;
  c1[k] = be1[k] - mu * a;
}

// ---------------------------------------------------------------------------
// Kernel 3: H2 = (H1*a1 + c1) @ W2 + b2 + BN2 stat partials.
// 1024 thr = 32 waves; tile 128(M) x 256(fullN); H1 read exactly once.
// ---------------------------------------------------------------------------
__global__ __launch_bounds__(1024) void gemm2_kernel(
    const float* __restrict__ H1, const float* __restrict__ W2,
    const float* __restrict__ b2, const float* __restrict__ a1,
    const float* __restrict__ c1, float* __restrict__ H2,
    float* __restrict__ cs, float* __restrict__ css)
{
  __shared__ __align__(16) char smem[(MT * 18 + N2 * SBT) * 4]; // 27.6 KB
  float* As = (float*)smem;                       // [128*18]
  float* Bs = (float*)(smem + MT * 18 * 4);       // [256*18] transposed [n][k]
  float* cP  = (float*)smem;                      // alias: [8*256]
  float* cPS = (float*)(smem + 8 * N2 * 4);       // alias: [8*256]

  const int tid  = threadIdx.x;
  const int wave = tid >> 5, lane = tid & 31;
  const int l16  = lane & 15, half = lane >> 4;
  const int waveM = wave & 7;        // rows waveM*16..+15
  const int waveN = wave >> 3;       // cols waveN*64..+63 (4 tiles)
  const int mBase = blockIdx.x * MT;

  v8f acc[4] = {};

  const bool stA = (tid < 512);
  const int sr = tid >> 2, sq = tid & 3;
  const int bk = tid >> 6, bnq = tid & 63;

  #pragma unroll 1
  for (int kb = 0; kb < N1; kb += 16) {
    __syncthreads();
    if (stA) {
      float4 av = *(const float4*)(a1 + kb + sq * 4);
      float4 cv = *(const float4*)(c1 + kb + sq * 4);
      float4 v  = *(const float4*)(H1 + (size_t)(mBase + sr) * N1 + kb + sq * 4);
      float* dst = &As[sr * 18 + sq * 4];
      dst[0] = v.x * av.x + cv.x;
      dst[1] = v.y * av.y + cv.y;
      dst[2] = v.z * av.z + cv.z;
      dst[3] = v.w * av.w + cv.w;
    }
    {
      float4 w = *(const float4*)(W2 + (size_t)(kb + bk) * N2 + bnq * 4);
      const int n0 = bnq * 4;
      Bs[(n0 + 0) * SBT + bk] = w.x;
      Bs[(n0 + 1) * SBT + bk] = w.y;
      Bs[(n0 + 2) * SBT + bk] = w.z;
      Bs[(n0 + 3) * SBT + bk] = w.w;
    }
    __syncthreads();
    const int mL = waveM * 16 + l16;
    #pragma unroll
    for (int kk = 0; kk < 4; ++kk) {
      const int kr = kk * 4 + half * 2;
      v2f a = *(const v2f*)(&As[mL * 18 + kr]);
      #pragma unroll
      for (int j = 0; j < 4; ++j) {
        const int col = waveN * 64 + j * 16 + l16;
        v2f b = *(const v2f*)(&Bs[col * SBT + kr]);
        acc[j] = wmma4(a, b, acc[j]);
      }
    }
  }
  __syncthreads();

  const int rowG = mBase + waveM * 16 + half * 8;
  #pragma unroll
  for (int j = 0; j < 4; ++j) {
    const int col = waveN * 64 + j * 16 + l16;
    float bv = b2[col];
    float s = 0.f, ss = 0.f;
    #pragma unroll
    for (int v = 0; v < 8; ++v) {
      float h = acc[j][v] + bv;
      H2[(size_t)(rowG + v) * N2 + col] = h;
      s += h; ss += h * h;
    }
    s  += __shfl_xor(s, 16, 32);
    ss += __shfl_xor(ss, 16, 32);
    if (half == 0) { cP[waveM * N2 + col] = s; cPS[waveM * N2 + col] = ss; }
  }
  __syncthreads();
  if (tid < N2) {
    float s = 0.f, ss = 0.f;
    #pragma unroll
    for (int w = 0; w < 8; ++w) { s += cP[w * N2 + tid]; ss += cPS[w * N2 + tid]; }
    cs [(size_t)blockIdx.x * N2 + tid] = s;
    css[(size_t)blockIdx.x * N2 + tid] = ss;
  }
}

// ---------------------------------------------------------------------------
// Kernel 4: BN2 stats -> alpha2; kconst partial per block (kc[0]+kc[1])
// ---------------------------------------------------------------------------
__global__ __launch_bounds__(128) void fin2_kernel(
    const float* __restrict__ cs, const float* __restrict__ css,
    const float* __restrict__ g2, const float* __restrict__ be2,
    float* __restrict__ alpha2, float* __restrict__ kc)
{
  __shared__ float red[128];
  int t = threadIdx.x;
  int c = blockIdx.x * 128 + t;
  float s = 0.f, ss = 0.f;
  for (int mt = 0; mt < NMT; ++mt) { s += cs[(size_t)mt * N2 + c]; ss += css[(size_t)mt * N2 + c]; }
  float mu  = s  * (1.0f / Bn);
  float var = ss * (1.0f / Bn) - mu * mu;
  float a = g2[c] / sqrtf(var + EPS);
  alpha2[c] = a;
  red[t] = be2[c] - mu * a;
  __syncthreads();
  for (int st = 64; st > 0; st >>= 1) {
    if (t < st) red[t] += red[t + st];
    __syncthreads();
  }
  if (t == 0) kc[blockIdx.x] = red[0];
}

// ---------------------------------------------------------------------------
// Kernel 5: out[b] = bias + fs[b] + kconst + dot(H2[b,:], alpha2)
// ---------------------------------------------------------------------------
__global__ __launch_bounds__(256) void final_kernel(
    const float* __restrict__ H2, const float* __restrict__ fs,
    const float* __restrict__ alpha2, const float* __restrict__ kc,
    const float* __restrict__ biasp, float* __restrict__ out)
{
  __shared__ __align__(16) float al[N2];
  int tid = threadIdx.x;
  al[tid] = alpha2[tid];
  __syncthreads();
  int wave = tid >> 5, lane = tid & 31;
  size_t row = (size_t)blockIdx.x * 8 + wave;
  const float4* h = (const float4*)(H2 + row * N2);
  float4 x0 = h[lane], x1 = h[lane + 32];
  float4 a0 = *(const float4*)(&al[lane * 4]);
  float4 a1v = *(const float4*)(&al[128 + lane * 4]);
  float d = x0.x * a0.x + x0.y * a0.y + x0.z * a0.z + x0.w * a0.w
          + x1.x * a1v.x + x1.y * a1v.y + x1.z * a1v.z + x1.w * a1v.w;
  #pragma unroll
  for (int off = 16; off > 0; off >>= 1) d += __shfl_xor(d, off, 32);
  if (lane == 0) out[row] = biasp[0] + fs[row] + (kc[0] + kc[1]) + d;
}

// ---------------------------------------------------------------------------
extern "C" void kernel_launch(void* const* d_in, const int* in_sizes, int n_in,
                              void* d_out, int out_size, void* d_ws, size_t ws_size,
                              hipStream_t stream) {
  const int*   Xi   = (const int*)  d_in[0];
  const float* Xv   = (const float*)d_in[1];
  const float* emb1 = (const float*)d_in[2];
  const float* emb2 = (const float*)d_in[3];
  const float* W1   = (const float*)d_in[4];
  const float* b1   = (const float*)d_in[5];
  const float* g1   = (const float*)d_in[6];
  const float* be1  = (const float*)d_in[7];
  const float* W2   = (const float*)d_in[8];
  const float* b2   = (const float*)d_in[9];
  const float* g2   = (const float*)d_in[10];
  const float* be2  = (const float*)d_in[11];
  const float* bias = (const float*)d_in[12];
  float* out = (float*)d_out;

  float* ws = (float*)d_ws;
  float* H1   = ws;                               // Bn*512
  float* H2   = H1   + (size_t)Bn * N1;           // Bn*256
  float* fs   = H2   + (size_t)Bn * N2;           // Bn
  float* cs1  = fs   + Bn;                        // 512*512
  float* css1 = cs1  + (size_t)NMT * N1;
  float* cs2  = css1 + (size_t)NMT * N1;          // 512*256
  float* css2 = cs2  + (size_t)NMT * N2;
  float* a1   = css2 + (size_t)NMT * N2;          // 512
  float* c1   = a1   + N1;                        // 512
  float* al2  = c1   + N1;                        // 256
  float* kc   = al2  + N2;                        // 2

  gemm1_fused_kernel<<<NMT, 1024, 0, stream>>>(Xi, Xv, emb1, emb2, W1, b1,
                                               H1, fs, cs1, css1);
  fin1_kernel<<<N1 / 128, 128, 0, stream>>>(cs1, css1, g1, be1, a1, c1);
  gemm2_kernel<<<NMT, 1024, 0, stream>>>(H1, W2, b2, a1, c1, H2, cs2, css2);
  fin2_kernel<<<N2 / 128, 128, 0, stream>>>(cs2, css2, g2, be2, al2, kc);
  final_kernel<<<Bn / 8, 256, 0, stream>>>(H2, fs, al2, kc, bias, out);
}